// GraphConvolutionWrapper_88536455840502
// MI455X (gfx1250) — compile-verified
//
#include <hip/hip_runtime.h>

// ---------------------------------------------------------------------------
// GCN layer for MI455X (gfx1250, wave32).
// FP32 WMMA (V_WMMA_F32_16X16X4_F32) for all GEMMs:
// the matrix work is far below the scatter-phase roofline
// (~3.8 GB of edge gather/scatter traffic => >=165us at 23.3 TB/s HBM).
// Each wave computes a 16 x (16*NT) output strip so one A-fragment ds_load
// pair feeds NT WMMA issues; K / leading dims are template constants so B
// loads use immediate offsets off one strength-reduced base pointer.
// ---------------------------------------------------------------------------

typedef __attribute__((ext_vector_type(2))) float v2f;
typedef __attribute__((ext_vector_type(8))) float v8f;

#define IN_F   256
#define EDGE_F 64
#define HALF_F 128
#define BASE_F 384
#define OUT_F  256
#define NEG_SLOPE 0.01f

// ---------------------------------------------------------------------------
__global__ __launch_bounds__(256) void zero_f32(float* __restrict__ p, long n) {
  long i = (long)blockIdx.x * blockDim.x + threadIdx.x;
  long stride = (long)gridDim.x * blockDim.x;
  for (; i < n; i += stride) p[i] = 0.0f;
}

// Scatter-add edge_attr by src node; count in-degree by dst.
// 16 threads per edge, 4 features each (float4 read).
__global__ __launch_bounds__(256) void edge_scatter(
    const int* __restrict__ ei, const float* __restrict__ ea,
    float* __restrict__ per_node, float* __restrict__ deg, int E) {
  long t = (long)blockIdx.x * blockDim.x + threadIdx.x;
  long total = (long)E * 16;
  if (t >= total) return;
  int e = (int)(t >> 4);
  int f = (int)(t & 15) * 4;
  int s = ei[e];
  int d = ei[E + e];
  const float4 v = *reinterpret_cast<const float4*>(ea + (long)e * EDGE_F + f);
  float* p = per_node + (long)s * EDGE_F + f;
  atomicAdd(p + 0, v.x);
  atomicAdd(p + 1, v.y);
  atomicAdd(p + 2, v.z);
  atomicAdd(p + 3, v.w);
  if ((t & 15) == 0) atomicAdd(deg + d, 1.0f);
}

// h[:, 0:256] = x   (float4 copies)
__global__ __launch_bounds__(256) void copy_x_to_h(
    const float* __restrict__ x, float* __restrict__ h, int N) {
  long t = (long)blockIdx.x * blockDim.x + threadIdx.x;
  long total = (long)N * (IN_F / 4);
  if (t >= total) return;
  int r = (int)(t >> 6);
  int c = (int)(t & 63) * 4;
  *reinterpret_cast<float4*>(h + (long)r * BASE_F + c) =
      *reinterpret_cast<const float4*>(x + (long)r * IN_F + c);
}

// ---------------------------------------------------------------------------
// FP32 WMMA GEMM: out[m0:m0+16, :] = A[m0:m0+16, :K] @ W[K, Nc] (+bias)(+act)
// Block = 128 threads = 4 waves. Each wave owns NT 16x16 tiles (a 16 x 16*NT
// strip); a block covers 64*NT output columns from one shared LDS A tile.
// The 16xK A tile is staged transposed in LDS so A-fragment reads (lane = M
// row) are conflict-free ds loads; the fragment is reused for NT WMMAs per
// k-step. W (<=384 KB) stays L2-resident; B reads are 64B/half-wave segments
// at compile-time immediate offsets. Requires: M % 16 == 0,
// Nc % (64*NT) == 0, K % 4 == 0, K <= 384.   act: 0 = none, 1 = leaky_relu.
// ---------------------------------------------------------------------------
template <int NT, int K, int LDA, int LDW, int LDO>
__global__ __launch_bounds__(128) void wmma_gemm_f32(
    const float* __restrict__ A, const float* __restrict__ W,
    const float* __restrict__ bias0, const float* __restrict__ bias1,
    float* __restrict__ out, int act) {
  __shared__ float As[K * 16];  // transposed: As[k*16 + row]

  const int tid = threadIdx.x;
  const int m0 = blockIdx.x * 16;

  // Stage A tile (coalesced global reads, transposed LDS writes).
#pragma unroll 4
  for (int i = tid; i < 16 * K; i += 128) {
    int r = i / K;
    int k = i - r * K;
    As[k * 16 + r] = A[(long)(m0 + r) * LDA + k];
  }
  __syncthreads();

  const int lane = tid & 31;
  const int wave = tid >> 5;
  const int nbase = blockIdx.y * (64 * NT) + wave * (16 * NT);
  const int mrow = lane & 15;            // A: M row, B/D: N col (within tile)
  const int kq = (lane >> 4) << 1;       // K sub-pair: 0 or 2

  v8f acc[NT];
#pragma unroll
  for (int j = 0; j < NT; ++j) acc[j] = {};

  // Per-lane B base: row (kq), column (nbase+mrow). All further offsets are
  // compile-time immediates; base advances by 4*LDW floats per k-step.
  const float* Wp = W + (long)kq * LDW + nbase + mrow;
  const float* Ap = As + kq * 16 + mrow;

  for (int k0 = 0; k0 < K; k0 += 4) {
    v2f a;
    // A fragment (ISA 7.12.2): lanes 0-15 hold K={0,1}, lanes 16-31 K={2,3}
    a[0] = Ap[0];
    a[1] = Ap[16];
#pragma unroll
    for (int j = 0; j < NT; ++j) {
      v2f b;  // B fragment: K rows striped across VGPRs, N across half-waves
      b[0] = Wp[j * 16];
      b[1] = Wp[j * 16 + LDW];
      acc[j] = __builtin_amdgcn_wmma_f32_16x16x4_f32(
          /*neg_a=*/false, a, /*neg_b=*/false, b,
          /*c_mod=*/(short)0, acc[j], /*reuse_a=*/false, /*reuse_b=*/false);
    }
    Wp += 4 * LDW;
    Ap += 4 * 16;
  }

  // D layout: VGPR i -> M = i (+8 for lanes 16-31), N = lane & 15.
  const int rbase = m0 + ((lane >> 4) << 3);
#pragma unroll
  for (int j = 0; j < NT; ++j) {
    const int col = nbase + j * 16 + mrow;
    float badd = 0.0f;
    if (bias0) badd += bias0[col];
    if (bias1) badd += bias1[col];
#pragma unroll
    for (int i = 0; i < 8; ++i) {
      float v = acc[j][i] + badd;
      if (act == 1) v = (v > 0.0f) ? v : v * NEG_SLOPE;
      out[(long)(rbase + i) * LDO + col] = v;
    }
  }
}

// out[i,:] += hw[i,:] / (deg[i] + 1)   (self-loop term, dinv^2 = 1/deg)
__global__ __launch_bounds__(256) void self_loop_add(
    const float* __restrict__ hw, const float* __restrict__ deg,
    float* __restrict__ out, int N) {
  long t = (long)blockIdx.x * blockDim.x + threadIdx.x;
  long total = (long)N * (OUT_F / 4);
  if (t >= total) return;
  int r = (int)(t >> 6);
  int c = (int)(t & 63) * 4;
  float inv = 1.0f / (deg[r] + 1.0f);
  float4 v = *reinterpret_cast<const float4*>(hw + (long)r * OUT_F + c);
  float4* o = reinterpret_cast<float4*>(out + (long)r * OUT_F + c);
  float4 w = *o;
  w.x += v.x * inv; w.y += v.y * inv; w.z += v.z * inv; w.w += v.w * inv;
  *o = w;
}

// out[dst,:] += rsqrt(deg_s) * rsqrt(deg_d) * hw[src,:]
// 64 threads per edge, 4 features each.
__global__ __launch_bounds__(256) void edge_aggregate(
    const int* __restrict__ ei, const float* __restrict__ hw,
    const float* __restrict__ deg, float* __restrict__ out, int E) {
  long t = (long)blockIdx.x * blockDim.x + threadIdx.x;
  long total = (long)E * (OUT_F / 4);
  if (t >= total) return;
  int e = (int)(t >> 6);
  int c = (int)(t & 63) * 4;
  int s = ei[e];
  int d = ei[E + e];
  float norm = rsqrtf(deg[s] + 1.0f) * rsqrtf(deg[d] + 1.0f);
  float4 v = *reinterpret_cast<const float4*>(hw + (long)s * OUT_F + c);
  float* p = out + (long)d * OUT_F + c;
  atomicAdd(p + 0, norm * v.x);
  atomicAdd(p + 1, norm * v.y);
  atomicAdd(p + 2, norm * v.z);
  atomicAdd(p + 3, norm * v.w);
}

__global__ __launch_bounds__(256) void relu_inplace(float* __restrict__ out, long n4) {
  long t = (long)blockIdx.x * blockDim.x + threadIdx.x;
  if (t >= n4) return;
  float4* p = reinterpret_cast<float4*>(out) + t;
  float4 v = *p;
  v.x = fmaxf(v.x, 0.0f);
  v.y = fmaxf(v.y, 0.0f);
  v.z = fmaxf(v.z, 0.0f);
  v.w = fmaxf(v.w, 0.0f);
  *p = v;
}

// ---------------------------------------------------------------------------
extern "C" void kernel_launch(void* const* d_in, const int* in_sizes, int n_in,
                              void* d_out, int out_size, void* d_ws, size_t ws_size,
                              hipStream_t stream) {
  const float* x      = (const float*)d_in[0];
  const int*   ei     = (const int*)d_in[1];   // [2, E]: row 0 = src, row 1 = dst
  const float* ea     = (const float*)d_in[2]; // [E, 64]
  // d_in[3] = batch (unused by reference output)
  const float* W_edge = (const float*)d_in[4]; // [64, 128]
  const float* b_edge = (const float*)d_in[5];
  const float* W_conv = (const float*)d_in[6]; // [384, 256]
  const float* b_conv = (const float*)d_in[7];
  const float* W_res  = (const float*)d_in[8]; // [384, 256]
  const float* b_res  = (const float*)d_in[9];

  const int N = in_sizes[0] / IN_F;    // 50000 (multiple of 16)
  const int E = in_sizes[2] / EDGE_F;  // 1,600,000

  // Workspace layout (floats): per_node[N*64] | deg[N] | h[N*384] | hw[N*256]
  float* ws       = (float*)d_ws;
  float* per_node = ws;
  float* deg      = per_node + (long)N * EDGE_F;
  float* h        = deg + N;
  float* hw       = h + (long)N * BASE_F;
  float* out      = (float*)d_out;

  // 1. zero accumulators (per_node + deg are contiguous)
  zero_f32<<<2048, 256, 0, stream>>>(ws, (long)N * (EDGE_F + 1));

  // 2. scatter edge_attr by src; in-degree by dst
  {
    long total = (long)E * 16;
    edge_scatter<<<(int)((total + 255) / 256), 256, 0, stream>>>(ei, ea, per_node, deg, E);
  }

  // 3a. h[:, 0:256] = x
  {
    long total = (long)N * (IN_F / 4);
    copy_x_to_h<<<(int)((total + 255) / 256), 256, 0, stream>>>(x, h, N);
  }

  // 3b. h[:, 256:384] = leaky_relu(per_node @ W_edge + b_edge)
  //     [WMMA, K=64; 4 waves x 32 cols = 128 cols per block]
  {
    dim3 g(N / 16, 1);
    wmma_gemm_f32<2, EDGE_F, EDGE_F, HALF_F, BASE_F>
        <<<g, 128, 0, stream>>>(per_node, W_edge, b_edge, nullptr,
                                h + IN_F, /*act=*/1);
  }

  // 4a. hw = h @ W_conv   [WMMA, K=384; 4 waves x 64 cols = 256 cols per block]
  {
    dim3 g(N / 16, 1);
    wmma_gemm_f32<4, BASE_F, BASE_F, OUT_F, OUT_F>
        <<<g, 128, 0, stream>>>(h, W_conv, nullptr, nullptr, hw, /*act=*/0);
  }

  // 4b. out = h @ W_res + b_conv + b_res
  {
    dim3 g(N / 16, 1);
    wmma_gemm_f32<4, BASE_F, BASE_F, OUT_F, OUT_F>
        <<<g, 128, 0, stream>>>(h, W_res, b_conv, b_res, out, /*act=*/0);
  }

  // 5. out += hw / deg  (self-loop, deg includes +1 self-loop)
  {
    long total = (long)N * (OUT_F / 4);
    self_loop_add<<<(int)((total + 255) / 256), 256, 0, stream>>>(hw, deg, out, N);
  }

  // 6. out[dst] += norm * hw[src]   (dominant phase: ~3.3 GB gather/scatter)
  {
    long total = (long)E * (OUT_F / 4);
    edge_aggregate<<<(int)((total + 255) / 256), 256, 0, stream>>>(ei, hw, deg, out, E);
  }

  // 7. ReLU in place
  {
    long n4 = (long)N * (OUT_F / 4);
    relu_inplace<<<(int)((n4 + 255) / 256), 256, 0, stream>>>(out, n4);
  }
}